// AttCNN4Weight_7911329759992
// MI455X (gfx1250) — compile-verified
//
#include <hip/hip_runtime.h>
#include <math.h>

// Problem dimensions (fixed by setup_inputs).
#define L_DIM 2048
#define B_DIM 64
#define C_DIM 512
#define Q_DIM 512
#define V_DIM 512
#define KW    3
#define NKER  (C_DIM * KW)   // 1536
#define T_CHUNK 64           // t-values per score block
#define L_SPLIT 16           // l-chunks for attend partials

typedef __attribute__((ext_vector_type(2))) float v2f;
typedef __attribute__((ext_vector_type(8))) float v8f;

// ---------------------------------------------------------------------------
// Kernel A: coef[B][C*KW] = q @ W^T + bias  using native f32 WMMA 16x16x4.
// One wave per 16x16 output tile; K loop of Q/4 = 128 steps.
// A-matrix (16x4, f32): lanes 0-15 -> M=lane, VGPR0=K0, VGPR1=K1;
//                       lanes 16-31 -> M=lane-16, VGPR0=K2, VGPR1=K3.
// B-matrix (4x16) mirrors: lanes give N, VGPR pair gives the K quad.
// C/D (16x16 f32): lanes 0-15 rows 0-7, lanes 16-31 rows 8-15.
// ---------------------------------------------------------------------------
__global__ __launch_bounds__(32) void coef_gemm(const float* __restrict__ q,
                                                const float* __restrict__ W,
                                                const float* __restrict__ bias,
                                                float* __restrict__ coef) {
  const int lane = threadIdx.x;
  const int half = lane >> 4;
  const int l15  = lane & 15;
  const int tm = blockIdx.y * 16;  // over B rows
  const int tn = blockIdx.x * 16;  // over C*KW cols
  const float* qrow = q + (size_t)(tm + l15) * Q_DIM + 2 * half;
  const float* wrow = W + (size_t)(tn + l15) * Q_DIM + 2 * half;  // W^T[k][n] = W[n][k]
  v8f acc = {};
  for (int k0 = 0; k0 < Q_DIM; k0 += 4) {
    v2f a, b;
    a.x = qrow[k0];  a.y = qrow[k0 + 1];
    b.x = wrow[k0];  b.y = wrow[k0 + 1];
    acc = __builtin_amdgcn_wmma_f32_16x16x4_f32(false, a, false, b,
                                                (short)0, acc, false, false);
  }
  const float bv = bias[tn + l15];
  const int rbase = tm + half * 8;
#pragma unroll
  for (int i = 0; i < 8; ++i)
    coef[(size_t)(rbase + i) * NKER + tn + l15] = acc[i] + bv;
}

// ---------------------------------------------------------------------------
// Kernel B: a_ij[t,b] = sum_w k[t+w-1,b,:] . ker[b,:,w]   (zero-padded ends)
// Each block: fixed b, 64 consecutive t.  Reads each needed k-row exactly
// once (2-row halo).  coef staged in LDS as [w][c] planes.
// s[u][w] = k[u,b,:].ker[:,w];  a[t] = s[t-1][0] + s[t][1] + s[t+1][2].
// ---------------------------------------------------------------------------
__global__ __launch_bounds__(256) void score_kernel(const float* __restrict__ k,
                                                    const float* __restrict__ coef,
                                                    float* __restrict__ a_out) {
  __shared__ float sCoef[NKER];            // [w*C_DIM + c]
  __shared__ float sS[(T_CHUNK + 2) * 4];  // s[row][w], stride 4
  const int b  = blockIdx.y;
  const int t0 = blockIdx.x * T_CHUNK;

  for (int i = threadIdx.x; i < NKER; i += blockDim.x) {
    const int c = i / KW;
    const int w = i - c * KW;
    sCoef[w * C_DIM + c] = coef[(size_t)b * NKER + i];
  }
  __syncthreads();

  const int lane = threadIdx.x & 31;
  const int wave = threadIdx.x >> 5;
  for (int idx = wave; idx < T_CHUNK + 2; idx += 8) {
    const int u = t0 - 1 + idx;
    float a0 = 0.f, a1 = 0.f, a2 = 0.f;
    if (u >= 0 && u < L_DIM) {               // uniform per-wave, no divergence
      const float* krow = k + ((size_t)u * B_DIM + b) * C_DIM;
      const int un = u + 8;
      if (un < L_DIM) {                      // prefetch this wave's next row
        const float* knext = k + ((size_t)un * B_DIM + b) * C_DIM;
        __builtin_prefetch(knext + lane * 16, 0, 1);  // 32 lanes x 64B = 2KB row
      }
#pragma unroll
      for (int j = 0; j < C_DIM / 32; ++j) {
        const int c = lane + j * 32;         // lane-contiguous: coalesced + bank-free
        const float kv = krow[c];
        a0 = fmaf(kv, sCoef[c], a0);
        a1 = fmaf(kv, sCoef[C_DIM + c], a1);
        a2 = fmaf(kv, sCoef[2 * C_DIM + c], a2);
      }
#pragma unroll
      for (int m = 16; m >= 1; m >>= 1) {    // wave32 reduction
        a0 += __shfl_xor(a0, m, 32);
        a1 += __shfl_xor(a1, m, 32);
        a2 += __shfl_xor(a2, m, 32);
      }
    }
    if (lane == 0) {
      sS[idx * 4 + 0] = a0;
      sS[idx * 4 + 1] = a1;
      sS[idx * 4 + 2] = a2;
    }
  }
  __syncthreads();

  if (threadIdx.x < T_CHUNK) {
    const int i = threadIdx.x;               // row u = t-1 lives at slot i
    const int t = t0 + i;
    const float a = sS[i * 4 + 0] + sS[(i + 1) * 4 + 1] + sS[(i + 2) * 4 + 2];
    a_out[(size_t)t * B_DIM + b] = a;
  }
}

// ---------------------------------------------------------------------------
// Kernel C: masked softmax over L for one column b per block.
// e = mask ? exp(a - max_masked) / sum_masked : 0
// ---------------------------------------------------------------------------
__global__ __launch_bounds__(256) void softmax_kernel(const float* __restrict__ a_in,
                                                      const int* __restrict__ k_mask,
                                                      float* __restrict__ e_out) {
  const int b = blockIdx.x;
  __shared__ float red[256];
  float vals[L_DIM / 256];
  int   msk[L_DIM / 256];
  float mx = -3.402823466e38f;
#pragma unroll
  for (int j = 0; j < L_DIM / 256; ++j) {
    const int t = threadIdx.x + j * 256;
    vals[j] = a_in[(size_t)t * B_DIM + b];
    msk[j]  = k_mask[(size_t)t * B_DIM + b];
    if (msk[j]) mx = fmaxf(mx, vals[j]);
  }
  red[threadIdx.x] = mx;
  __syncthreads();
  for (int s = 128; s >= 1; s >>= 1) {
    if (threadIdx.x < s) red[threadIdx.x] = fmaxf(red[threadIdx.x], red[threadIdx.x + s]);
    __syncthreads();
  }
  mx = red[0];
  __syncthreads();

  float ex[L_DIM / 256];
  float sum = 0.f;
#pragma unroll
  for (int j = 0; j < L_DIM / 256; ++j) {
    ex[j] = msk[j] ? __expf(vals[j] - mx) : 0.f;
    sum += ex[j];
  }
  red[threadIdx.x] = sum;
  __syncthreads();
  for (int s = 128; s >= 1; s >>= 1) {
    if (threadIdx.x < s) red[threadIdx.x] += red[threadIdx.x + s];
    __syncthreads();
  }
  const float inv = 1.f / red[0];
#pragma unroll
  for (int j = 0; j < L_DIM / 256; ++j) {
    const int t = threadIdx.x + j * 256;
    e_out[(size_t)t * B_DIM + b] = ex[j] * inv;
  }
}

// ---------------------------------------------------------------------------
// Kernel D: partial attend over an l-chunk: streams v exactly once.
// block = (chunk, b); 256 threads cover the 512 v-lanes (2 each).
// ---------------------------------------------------------------------------
__global__ __launch_bounds__(256) void attend_partial(const float* __restrict__ v,
                                                      const float* __restrict__ e,
                                                      float* __restrict__ partial) {
  const int b     = blockIdx.y;
  const int chunk = blockIdx.x;
  const int l0    = chunk * (L_DIM / L_SPLIT);
  const int tid   = threadIdx.x;
  const float* vbase = v + ((size_t)l0 * B_DIM + b) * V_DIM;
  float acc0 = 0.f, acc1 = 0.f;
  for (int l = 0; l < L_DIM / L_SPLIT; ++l) {
    const float w = e[(size_t)(l0 + l) * B_DIM + b];         // uniform -> scalar path
    const float* vr = vbase + (size_t)l * B_DIM * V_DIM;
    __builtin_prefetch(vr + (size_t)B_DIM * V_DIM + tid * 2, 0, 1);  // next 2KB row
    acc0 = fmaf(w, vr[tid], acc0);
    acc1 = fmaf(w, vr[tid + 256], acc1);
  }
  float* p = partial + ((size_t)(chunk * B_DIM + b)) * V_DIM;
  p[tid]       = acc0;
  p[tid + 256] = acc1;
}

// Kernel E: deterministic reduction of the L_SPLIT partials.
__global__ __launch_bounds__(256) void attend_reduce(const float* __restrict__ partial,
                                                     float* __restrict__ attend_out) {
  const int idx = blockIdx.x * 256 + threadIdx.x;  // over B*V = 32768
  const int b = idx / V_DIM, vj = idx - b * V_DIM;
  float s = 0.f;
#pragma unroll
  for (int c = 0; c < L_SPLIT; ++c)
    s += partial[((size_t)(c * B_DIM + b)) * V_DIM + vj];
  attend_out[idx] = s;
}

extern "C" void kernel_launch(void* const* d_in, const int* in_sizes, int n_in,
                              void* d_out, int out_size, void* d_ws, size_t ws_size,
                              hipStream_t stream) {
  (void)in_sizes; (void)n_in; (void)out_size; (void)ws_size;
  const float* q      = (const float*)d_in[0];
  const float* k      = (const float*)d_in[1];
  const float* v      = (const float*)d_in[2];
  const int*   k_mask = (const int*)d_in[3];
  const float* W      = (const float*)d_in[4];
  const float* bias   = (const float*)d_in[5];

  float* a_ij   = (float*)d_out;                       // [L, B]
  float* e_ij   = a_ij + (size_t)L_DIM * B_DIM;        // [L, B]
  float* attend = e_ij + (size_t)L_DIM * B_DIM;        // [B, V]

  float* coef    = (float*)d_ws;                       // [B, C*KW]   (384 KB)
  float* partial = coef + (size_t)B_DIM * NKER;        // [L_SPLIT, B, V] (2 MB)

  coef_gemm<<<dim3(NKER / 16, B_DIM / 16), 32, 0, stream>>>(q, W, bias, coef);
  score_kernel<<<dim3(L_DIM / T_CHUNK, B_DIM), 256, 0, stream>>>(k, coef, a_ij);
  softmax_kernel<<<B_DIM, 256, 0, stream>>>(a_ij, k_mask, e_ij);
  attend_partial<<<dim3(L_SPLIT, B_DIM), 256, 0, stream>>>(v, e_ij, partial);
  attend_reduce<<<(B_DIM * V_DIM) / 256, 256, 0, stream>>>(partial, attend);
}